// Word_alignment_65618510348774
// MI455X (gfx1250) — compile-verified
//
#include <hip/hip_runtime.h>
#include <hip/hip_bf16.h>

typedef __attribute__((ext_vector_type(2))) float v2f;
typedef __attribute__((ext_vector_type(8))) float v8f;
typedef __attribute__((ext_vector_type(4))) unsigned int su32x4;
typedef __attribute__((ext_vector_type(8))) unsigned int su32x8;

#define BATCH 8
#define S1    2048
#define S2    2048
#define DIM   1024
#define MTOT  (BATCH * S1)

#define KB    32   // K-chunk (floats) staged per TDM transfer
#define ST    36   // LDS row stride = KB + 4 pad dwords (TDM pad_interval=4, pad_amount=3)

#define WMMA_F32(a, b, c) \
  __builtin_amdgcn_wmma_f32_16x16x4_f32(false, (a), false, (b), (short)0, (c), false, false)

// Low 32 bits of a flat LDS pointer = byte offset within the workgroup's LDS.
__device__ __forceinline__ unsigned lds_off(const void* p) {
  return (unsigned)(unsigned long long)p;
}

// ---------------------------------------------------------------------------
// Tensor Data Mover: async 2D tile (tile_x dwords wide, tile_y rows) from a
// row-major fp32 matrix (row stride `stride0` elements) into LDS at lds_addr,
// padding +4 dwords after every 32 dwords -> LDS row stride = 36 floats.
// D# packing per CDNA5 ISA 8.3/8.4 (group0 128b, group1 256b, groups2/3 zero).
// Completion tracked with TENSORcnt.
// ---------------------------------------------------------------------------
__device__ __forceinline__ void tdm_load_2d(unsigned lds_addr, const float* gptr,
                                            unsigned tile_x, unsigned tile_y,
                                            unsigned stride0) {
  unsigned long long ga = (unsigned long long)gptr;
  su32x4 g0;
  g0.x = 1u;                                              // count=1, user descriptor
  g0.y = lds_addr;                                        // lds_addr [63:32]
  g0.z = (unsigned)(ga & 0xffffffffu);                    // global_addr [95:64]
  g0.w = ((unsigned)(ga >> 32) & 0x01ffffffu) | (2u << 30); // addr[56:32] | type=2
  su32x8 g1;
  g1.s0 = (2u << 16)            // data_size = 4B
        | (1u << 20)            // pad_enable
        | (4u << 22)            // pad_interval: 2^(4+1)=32 dwords
        | (3u << 25);           // pad_amount: 3+1=4 dwords
  g1.s1 = (tile_x & 0xffffu) << 16;                       // tensor_dim0[15:0]
  g1.s2 = (tile_x >> 16) | ((tile_y & 0xffffu) << 16);    // dim0 hi | tensor_dim1 lo
  g1.s3 = (tile_y >> 16) | ((tile_x & 0xffffu) << 16);    // dim1 hi | tile_dim0
  g1.s4 = tile_y & 0xffffu;                               // tile_dim1 (tile_dim2=0)
  g1.s5 = stride0;                                        // tensor_dim0_stride[31:0]
  g1.s6 = 0u;
  g1.s7 = 0u;
  su32x4 gz = {0u, 0u, 0u, 0u};
  asm volatile("tensor_load_to_lds %0, %1, %2, %3"
               :: "s"(g0), "s"(g1), "s"(gz), "s"(gz)
               : "memory");
}

// ---------------------------------------------------------------------------
// Kernel 1: out1[M=16384, N=1024] = input1[M,1024] @ weight[N,1024]^T (fp32 WMMA)
// 256 thr / 8 waves, tile 128(M) x 64(N); wave w -> 32x32 subtile.
// K-chunks of 32 staged by TDM, double-buffered (DMA(c+1) overlaps WMMA(c)).
// ---------------------------------------------------------------------------
#define K1_MT 128
#define K1_NT 64

__global__ __launch_bounds__(256) void k1_linear_wmma(const float* __restrict__ A,
                                                      const float* __restrict__ W,
                                                      float* __restrict__ C) {
  __shared__ __align__(16) float As[2][K1_MT * ST];
  __shared__ __align__(16) float Ws[2][K1_NT * ST];

  const int tid  = threadIdx.x;
  const int wave = tid >> 5;
  const int lane = tid & 31;
  const int ln   = lane & 15;
  const int ko   = (lane < 16) ? 0 : 2;
  const int mbase = blockIdx.y * K1_MT;
  const int nbase = blockIdx.x * K1_NT;
  const int mo = (wave >> 1) * 32;
  const int no = (wave & 1) * 32;

  const float* Abase = A + (size_t)mbase * DIM;
  const float* Wbase = W + (size_t)nbase * DIM;

  v8f acc[2][2];
#pragma unroll
  for (int i = 0; i < 2; ++i)
#pragma unroll
    for (int j = 0; j < 2; ++j) acc[i][j] = {};

  if (wave == 0) {   // prologue: fill buffer 0
    tdm_load_2d(lds_off(&As[0][0]), Abase, KB, K1_MT, DIM);
    tdm_load_2d(lds_off(&Ws[0][0]), Wbase, KB, K1_NT, DIM);
    __builtin_amdgcn_s_wait_tensorcnt(0);
  }
  __syncthreads();

  const int NCH = DIM / KB;
  for (int c = 0; c < NCH; ++c) {
    const int p = c & 1;
    if (wave == 0 && c + 1 < NCH) {   // prefetch next chunk into other buffer
      tdm_load_2d(lds_off(&As[p ^ 1][0]), Abase + (c + 1) * KB, KB, K1_MT, DIM);
      tdm_load_2d(lds_off(&Ws[p ^ 1][0]), Wbase + (c + 1) * KB, KB, K1_NT, DIM);
    }
    const float* as = As[p];
    const float* ws = Ws[p];
#pragma unroll
    for (int kk = 0; kk < KB; kk += 4) {
      v2f a0 = *(const v2f*)&as[(mo +      ln) * ST + kk + ko];
      v2f a1 = *(const v2f*)&as[(mo + 16 + ln) * ST + kk + ko];
      v2f b0 = *(const v2f*)&ws[(no +      ln) * ST + kk + ko];
      v2f b1 = *(const v2f*)&ws[(no + 16 + ln) * ST + kk + ko];
      acc[0][0] = WMMA_F32(a0, b0, acc[0][0]);
      acc[0][1] = WMMA_F32(a0, b1, acc[0][1]);
      acc[1][0] = WMMA_F32(a1, b0, acc[1][0]);
      acc[1][1] = WMMA_F32(a1, b1, acc[1][1]);
    }
    if (wave == 0 && c + 1 < NCH) __builtin_amdgcn_s_wait_tensorcnt(0);
    __syncthreads();
  }

  const int rbase = (lane < 16) ? 0 : 8;
#pragma unroll
  for (int mi = 0; mi < 2; ++mi)
#pragma unroll
    for (int ni = 0; ni < 2; ++ni)
#pragma unroll
      for (int r = 0; r < 8; ++r) {
        int row = mbase + mo + mi * 16 + rbase + r;
        int col = nbase + no + ni * 16 + ln;
        C[(size_t)row * DIM + col] = acc[mi][ni][r];
      }
}

// ---------------------------------------------------------------------------
// Kernel 2: fused scores + mask + online softmax(over s) + satt GEMM.
// Workgroup: (batch b, 32 t-cols), 256 thr / 8 waves; s-blocks of 64.
//   Phase A: S[64s x 32t] via WMMA, out1/in2 K-chunks staged by TDM (dbl-buf)
//   Phase B: threads 0..31: online max/sum per t, P^T -> LDS
//   Phase C: rescale acc, acc[t,d] += P^T x in1 (wave w owns d in [128w,128w+128))
// ---------------------------------------------------------------------------
#define TB     32
#define SBK    64
#define SL_ST  33
#define PT_ST  66

__global__ __launch_bounds__(256) void k2_flash_attn(const float* __restrict__ out1,
                                                     const float* __restrict__ in2,
                                                     const float* __restrict__ in1,
                                                     const unsigned char* __restrict__ mask,
                                                     float* __restrict__ out) {
  __shared__ __align__(16) float As2[2][SBK * ST];  // out1 chunk [64][36] x2
  __shared__ __align__(16) float Bs2[2][TB * ST];   // in2 chunk  [32][36] x2
  __shared__ __align__(16) float Sl[SBK * SL_ST];   // scores     [64][33]
  __shared__ __align__(16) float Ptl[TB * PT_ST];   // P^T        [32][66]
  __shared__ float scl[TB];
  __shared__ float lrow[TB];

  const int tid  = threadIdx.x;
  const int wave = tid >> 5;
  const int lane = tid & 31;
  const int ln   = lane & 15;
  const int hb   = (lane < 16) ? 0 : 1;
  const int ko   = hb * 2;
  const int rb   = hb * 8;

  const int b  = blockIdx.y;
  const int t0 = blockIdx.x * TB;

  const int si = wave >> 1;    // Phase A: s subtile 0..3
  const int ti = wave & 1;     // Phase A: t subtile 0..1
  const int d0 = wave * 128;   // Phase C: d range

  const float* out1b = out1 + (size_t)b * S1 * DIM;
  const float* in2b  = in2  + (size_t)b * S2 * DIM;
  const float* in1b  = in1  + (size_t)b * S1 * DIM;
  const unsigned char* maskb = mask + (size_t)b * S1;

  v8f acc[2][8];
#pragma unroll
  for (int m = 0; m < 2; ++m)
#pragma unroll
    for (int n = 0; n < 8; ++n) acc[m][n] = {};
  float m_t = -1e30f, l_t = 0.0f;

  const int NCH = DIM / KB;
  for (int s0 = 0; s0 < S1; s0 += SBK) {
    // ------------- Phase A: score subtile -------------
    v8f sacc = {};
    if (wave == 0) {
      tdm_load_2d(lds_off(&As2[0][0]), out1b + (size_t)s0 * DIM, KB, SBK, DIM);
      tdm_load_2d(lds_off(&Bs2[0][0]), in2b + (size_t)t0 * DIM, KB, TB, DIM);
      __builtin_amdgcn_s_wait_tensorcnt(0);
    }
    __syncthreads();
    for (int c = 0; c < NCH; ++c) {
      const int p = c & 1;
      if (wave == 0 && c + 1 < NCH) {
        tdm_load_2d(lds_off(&As2[p ^ 1][0]), out1b + (size_t)s0 * DIM + (c + 1) * KB,
                    KB, SBK, DIM);
        tdm_load_2d(lds_off(&Bs2[p ^ 1][0]), in2b + (size_t)t0 * DIM + (c + 1) * KB,
                    KB, TB, DIM);
      }
      const float* as = As2[p];
      const float* bs = Bs2[p];
#pragma unroll
      for (int kk = 0; kk < KB; kk += 4) {
        v2f a  = *(const v2f*)&as[(si * 16 + ln) * ST + kk + ko];
        v2f bb = *(const v2f*)&bs[(ti * 16 + ln) * ST + kk + ko];
        sacc = WMMA_F32(a, bb, sacc);
      }
      if (wave == 0 && c + 1 < NCH) __builtin_amdgcn_s_wait_tensorcnt(0);
      __syncthreads();
    }
    // masked scores -> Sl
#pragma unroll
    for (int r = 0; r < 8; ++r) {
      const int sl = si * 16 + rb + r;
      const int tl = ti * 16 + ln;
      float v = maskb[s0 + sl] ? sacc[r] : -100.0f;
      Sl[sl * SL_ST + tl] = v;
    }
    __syncthreads();
    // ------------- Phase B: online softmax stats (thread t = tid < 32) -------------
    if (tid < TB) {
      float mnew = m_t;
#pragma unroll 8
      for (int s = 0; s < SBK; ++s) mnew = fmaxf(mnew, Sl[s * SL_ST + tid]);
      const float sc = __expf(m_t - mnew);
      float psum = 0.0f;
#pragma unroll 8
      for (int s = 0; s < SBK; ++s) {
        float p = __expf(Sl[s * SL_ST + tid] - mnew);
        Ptl[tid * PT_ST + s] = p;
        psum += p;
      }
      l_t = l_t * sc + psum;
      m_t = mnew;
      scl[tid] = sc;
    }
    __syncthreads();
    // ------------- Phase C: rescale acc + accumulate P^T x in1 -------------
    v2f av[2][16];
#pragma unroll
    for (int ms = 0; ms < 2; ++ms)
#pragma unroll
      for (int ks = 0; ks < 16; ++ks)
        av[ms][ks] = *(const v2f*)&Ptl[(ms * 16 + ln) * PT_ST + ks * 4 + ko];
#pragma unroll
    for (int ms = 0; ms < 2; ++ms)
#pragma unroll
      for (int r = 0; r < 8; ++r) {
        const float sc = scl[ms * 16 + rb + r];
#pragma unroll
        for (int n = 0; n < 8; ++n) acc[ms][n][r] *= sc;
      }
#pragma unroll 2
    for (int n = 0; n < 8; ++n) {
      const float* bp = in1b + (size_t)s0 * DIM + d0 + n * 16 + ln;
#pragma unroll
      for (int ks = 0; ks < 16; ++ks) {
        const int krow = ks * 4 + ko;
        v2f bv;
        bv.x = bp[(size_t)krow * DIM];
        bv.y = bp[(size_t)(krow + 1) * DIM];
        acc[0][n] = WMMA_F32(av[0][ks], bv, acc[0][n]);
        acc[1][n] = WMMA_F32(av[1][ks], bv, acc[1][n]);
      }
    }
  }

  // ------------- Finalize: divide by l, store -------------
  if (tid < TB) lrow[tid] = l_t;
  __syncthreads();
#pragma unroll
  for (int ms = 0; ms < 2; ++ms)
#pragma unroll
    for (int r = 0; r < 8; ++r) {
      const int tg  = t0 + ms * 16 + rb + r;
      const float inv = 1.0f / lrow[ms * 16 + rb + r];
      float* op = out + ((size_t)b * S2 + tg) * DIM + d0 + ln;
#pragma unroll
      for (int n = 0; n < 8; ++n) op[n * 16] = acc[ms][n][r] * inv;
    }
}

// ---------------------------------------------------------------------------
extern "C" void kernel_launch(void* const* d_in, const int* in_sizes, int n_in,
                              void* d_out, int out_size, void* d_ws, size_t ws_size,
                              hipStream_t stream) {
  const float* input1 = (const float*)d_in[0];          // [8,2048,1024] f32
  const float* input2 = (const float*)d_in[1];          // [8,2048,1024] f32
  const unsigned char* input_mask = (const unsigned char*)d_in[2];  // [8,2048] bool
  const float* weight = (const float*)d_in[3];          // [1024,1024] f32
  float* out  = (float*)d_out;                          // [8,2048,1024] f32
  float* out1 = (float*)d_ws;                           // 64 MB scratch: input1 @ W^T

  (void)in_sizes; (void)n_in; (void)out_size; (void)ws_size;

  k1_linear_wmma<<<dim3(DIM / K1_NT, MTOT / K1_MT), 256, 0, stream>>>(input1, weight, out1);
  k2_flash_attn<<<dim3(S2 / TB, BATCH), 256, 0, stream>>>(out1, input2, input1, input_mask, out);
}